// nearAttention_21663815041049
// MI455X (gfx1250) — compile-verified
//
#include <hip/hip_runtime.h>
#include <hip/hip_bf16.h>
#include <math.h>

// Problem constants (from reference)
#define B_   4
#define L_   1024
#define D_   768
#define H_   6
#define DH_  128
#define EPS_ 1e-5f

typedef float v2f __attribute__((ext_vector_type(2)));
typedef float v8f __attribute__((ext_vector_type(8)));

// Workspace layout (in floats)
#define WS_Q2     0                          // B*H*L
#define WS_K2     (B_*H_*L_)                 // B*H*L
#define WS_STATS  (2*B_*H_*L_)               // 2*H  (sum, sumsq per head)
#define WS_MEAN   (2*B_*H_*L_ + 2*H_)        // H
#define WS_INVSTD (2*B_*H_*L_ + 3*H_)        // H
#define WS_WSUM   (2*B_*H_*L_ + 4*H_)        // B*H*L (column sums)

// ---------------------------------------------------------------------------
// Core tile GEMM: one workgroup (8 waves) owns a 16 x 1024 block of co for one
// (b, h).  Each wave computes 8 tiles of 16x16 via V_WMMA_F32_16X16X4_F32.
// Software-pipelined: fragments for k-step n+1 are loaded while the 8 WMMAs of
// k-step n issue, so each step pays one s_wait_loadcnt for 9 batched loads.
// Addressing: lane-invariant SGPR base + 32-bit per-lane offset + imm (GVS).
// A fragment (16x4 f32): lane l holds row (l&15), k = kk + 2*(l>>4) + {0,1}
// B fragment (4x16 f32): lane l holds K-matrix row n (as col), same k pairing
// ---------------------------------------------------------------------------
__device__ __forceinline__ void qk_tiles(const float* __restrict__ Q,
                                         const float* __restrict__ K,
                                         int b, int h, int i0, int wave, int lane,
                                         v8f acc[8]) {
  const int hl = lane >> 4;
  const int ln = lane & 15;
  // lane-invariant bases (SGPR), per-lane 32-bit element offsets (VGPR)
  const float* qbase = Q + (size_t)b * (L_ * D_) + h * DH_;
  const float* kbase = K + (size_t)b * (L_ * D_) + h * DH_;
  const int qoff = (i0 + ln) * D_ + 2 * hl;
  int koff[8];
#pragma unroll
  for (int t = 0; t < 8; ++t) {
    koff[t] = ((wave * 8 + t) * 16 + ln) * D_ + 2 * hl;
    v8f z = {};
    acc[t] = z;
  }

  // prologue: k-step 0 fragments
  v2f a0 = *(const v2f*)(qbase + qoff);
  v2f b0[8];
#pragma unroll
  for (int t = 0; t < 8; ++t) b0[t] = *(const v2f*)(kbase + koff[t]);

#pragma unroll
  for (int kk = 0; kk < DH_ - 4; kk += 4) {
    // issue next k-step's 9 loads before consuming current fragments
    v2f a1 = *(const v2f*)(qbase + qoff + (kk + 4));
    v2f b1[8];
#pragma unroll
    for (int t = 0; t < 8; ++t)
      b1[t] = *(const v2f*)(kbase + koff[t] + (kk + 4));
#pragma unroll
    for (int t = 0; t < 8; ++t)
      acc[t] = __builtin_amdgcn_wmma_f32_16x16x4_f32(
          false, a0, false, b0[t], (short)0, acc[t], false, false);
    a0 = a1;
#pragma unroll
    for (int t = 0; t < 8; ++t) b0[t] = b1[t];
  }
  // epilogue: last k-step
#pragma unroll
  for (int t = 0; t < 8; ++t)
    acc[t] = __builtin_amdgcn_wmma_f32_16x16x4_f32(
        false, a0, false, b0[t], (short)0, acc[t], false, false);
}

// ---------------------------------------------------------------------------
// Kernel 0: per-(b,h,l) squared norms of Q and K rows
// ---------------------------------------------------------------------------
__global__ __launch_bounds__(256) void sq_kernel(const float* __restrict__ Q,
                                                 const float* __restrict__ K,
                                                 float* __restrict__ q2,
                                                 float* __restrict__ k2) {
  int t = blockIdx.x * blockDim.x + threadIdx.x;
  if (t >= B_ * H_ * L_) return;
  int l  = t % L_;
  int hh = (t / L_) % H_;
  int bb = t / (L_ * H_);
  const float* qp = Q + (size_t)(bb * L_ + l) * D_ + hh * DH_;
  const float* kp = K + (size_t)(bb * L_ + l) * D_ + hh * DH_;
  float sq = 0.f, sk = 0.f;
#pragma unroll 8
  for (int d = 0; d < DH_; ++d) {
    float a = qp[d]; sq = fmaf(a, a, sq);
    float c = kp[d]; sk = fmaf(c, c, sk);
  }
  q2[t] = sq;
  k2[t] = sk;
}

// ---------------------------------------------------------------------------
// Kernel 1: BN1 statistics pass — compute co tiles, reduce sum / sumsq per
// head.  co never touches memory here.
// ---------------------------------------------------------------------------
__global__ __launch_bounds__(256) void stats_kernel(const float* __restrict__ Q,
                                                    const float* __restrict__ K,
                                                    const float* __restrict__ q2,
                                                    const float* __restrict__ k2,
                                                    float* __restrict__ stats) {
  const int ib = blockIdx.x & 63;            // L/16 = 64 row-blocks
  const int h  = (blockIdx.x >> 6) % H_;
  const int b  = blockIdx.x / (64 * H_);
  const int wave = threadIdx.x >> 5;
  const int lane = threadIdx.x & 31;
  const int hl = lane >> 4, ln = lane & 15;
  const int i0 = ib * 16;

  v8f acc[8];
  qk_tiles(Q, K, b, h, i0, wave, lane, acc);

  const float* q2p = q2 + (size_t)(b * H_ + h) * L_;
  const float* k2p = k2 + (size_t)(b * H_ + h) * L_;

  float q2v[8];
#pragma unroll
  for (int r = 0; r < 8; ++r) q2v[r] = q2p[i0 + r + 8 * hl];

  float s1 = 0.f, s2 = 0.f;
#pragma unroll
  for (int t = 0; t < 8; ++t) {
    int j = (wave * 8 + t) * 16 + ln;
    float k2j = k2p[j];
#pragma unroll
    for (int r = 0; r < 8; ++r) {
      float co = 2.0f * acc[t][r] - q2v[r] - k2j;   // = -(q2+k2-2qk)
      s1 += co;
      s2 = fmaf(co, co, s2);
    }
  }
  __shared__ float a1[256];
  __shared__ float a2[256];
  a1[threadIdx.x] = s1;
  a2[threadIdx.x] = s2;
  __syncthreads();
  for (int s = 128; s > 0; s >>= 1) {
    if (threadIdx.x < s) {
      a1[threadIdx.x] += a1[threadIdx.x + s];
      a2[threadIdx.x] += a2[threadIdx.x + s];
    }
    __syncthreads();
  }
  if (threadIdx.x == 0) {
    atomicAdd(&stats[2 * h + 0], a1[0]);
    atomicAdd(&stats[2 * h + 1], a2[0]);
  }
}

// ---------------------------------------------------------------------------
// Kernel 1b: finalize per-head mean / inv-std
// ---------------------------------------------------------------------------
__global__ void stats_final(const float* __restrict__ stats,
                            float* __restrict__ meanv,
                            float* __restrict__ invs) {
  int h = threadIdx.x;
  if (h < H_) {
    const float cnt = (float)B_ * (float)L_ * (float)L_;
    float mean = stats[2 * h] / cnt;
    float var  = stats[2 * h + 1] / cnt - mean * mean;
    meanv[h] = mean;
    invs[h]  = rsqrtf(fmaxf(var, 0.f) + EPS_);
  }
}

// ---------------------------------------------------------------------------
// Kernel 2: fused GEMM -> BN1 -> mask -> row softmax -> store co + column sums.
// One workgroup = 16 full rows (entire softmax axis in registers + LDS).
// co stores are non-temporal: written once, never re-read on device; keeps the
// L2 free for the Q/K panels that all 64 row-blocks per (b,h) re-read.
// ---------------------------------------------------------------------------
__global__ __launch_bounds__(256) void co_softmax_kernel(
    const float* __restrict__ Q, const float* __restrict__ K,
    const float* __restrict__ q2, const float* __restrict__ k2,
    const float* __restrict__ meanv, const float* __restrict__ invs,
    const float* __restrict__ g1, const float* __restrict__ b1,
    const unsigned char* __restrict__ bx,
    float* __restrict__ co_out, float* __restrict__ wsum) {
  const int ib = blockIdx.x & 63;
  const int h  = (blockIdx.x >> 6) % H_;
  const int b  = blockIdx.x / (64 * H_);
  const int wave = threadIdx.x >> 5;
  const int lane = threadIdx.x & 31;
  const int hl = lane >> 4, ln = lane & 15;
  const int i0 = ib * 16;

  v8f acc[8];
  qk_tiles(Q, K, b, h, i0, wave, lane, acc);

  const float* q2p = q2 + (size_t)(b * H_ + h) * L_;
  const float* k2p = k2 + (size_t)(b * H_ + h) * L_;
  const unsigned char* bxb = bx + b * L_;
  const float mean = meanv[h];
  const float istd = invs[h];
  const float ga = g1[h];
  const float be = b1[h];

  float q2v[8];
  bool  vI[8];
#pragma unroll
  for (int r = 0; r < 8; ++r) {
    int m = i0 + r + 8 * hl;
    q2v[r] = q2p[m];
    vI[r]  = (bxb[m] == 0);
  }

  float x[8][8];  // [tile][reg] : this lane's 64 elements of the 16x1024 block
#pragma unroll
  for (int t = 0; t < 8; ++t) {
    int j = (wave * 8 + t) * 16 + ln;
    float k2j = k2p[j];
    bool vJ = (bxb[j] == 0);
#pragma unroll
    for (int r = 0; r < 8; ++r) {
      float raw = 2.0f * acc[t][r] - q2v[r] - k2j;
      float xn = fmaf((raw - mean) * istd, ga, be);
      x[t][r] = (vI[r] && vJ) ? xn : 0.0f;   // masked_fill before softmax
    }
  }

  __shared__ float red[16][8];   // [row-in-block][wave]
  __shared__ float rowm[16];
  __shared__ float rows[16];

  // --- row max across 1024 columns ---
#pragma unroll
  for (int r = 0; r < 8; ++r) {
    float m = x[0][r];
#pragma unroll
    for (int t = 1; t < 8; ++t) m = fmaxf(m, x[t][r]);
    // reduce across the 16 lanes of this half-wave (same row)
    for (int off = 1; off < 16; off <<= 1) m = fmaxf(m, __shfl_xor(m, off, 32));
    if (ln == 0) red[r + 8 * hl][wave] = m;
  }
  __syncthreads();
  if (threadIdx.x < 16) {
    float m = red[threadIdx.x][0];
    for (int wv = 1; wv < 8; ++wv) m = fmaxf(m, red[threadIdx.x][wv]);
    rowm[threadIdx.x] = m;
  }
  __syncthreads();
  float rm[8];
#pragma unroll
  for (int r = 0; r < 8; ++r) rm[r] = rowm[r + 8 * hl];

  // --- exp and row sum ---
#pragma unroll
  for (int r = 0; r < 8; ++r) {
    float s = 0.f;
#pragma unroll
    for (int t = 0; t < 8; ++t) {
      x[t][r] = __expf(x[t][r] - rm[r]);
      s += x[t][r];
    }
    for (int off = 1; off < 16; off <<= 1) s += __shfl_xor(s, off, 32);
    if (ln == 0) red[r + 8 * hl][wave] = s;
  }
  __syncthreads();
  if (threadIdx.x < 16) {
    float s = 0.f;
    for (int wv = 0; wv < 8; ++wv) s += red[threadIdx.x][wv];
    rows[threadIdx.x] = s;
  }
  __syncthreads();
  float rinv[8];
#pragma unroll
  for (int r = 0; r < 8; ++r) rinv[r] = 1.0f / rows[r + 8 * hl];  // sum >= 1

  // --- write softmaxed co (the single 100 MB HBM write) + column sums ---
  float* cobh = co_out + (size_t)(b * H_ + h) * L_ * L_;
  float* wsp  = wsum + (size_t)(b * H_ + h) * L_;
#pragma unroll
  for (int t = 0; t < 8; ++t) {
    int j = (wave * 8 + t) * 16 + ln;
    float cs = 0.f;
#pragma unroll
    for (int r = 0; r < 8; ++r) {
      float p = x[t][r] * rinv[r];
      cs += p;
      __builtin_nontemporal_store(p, &cobh[(size_t)(i0 + r + 8 * hl) * L_ + j]);
    }
    cs += __shfl_xor(cs, 16, 32);            // combine the two 8-row halves
    if (hl == 0) atomicAdd(&wsp[j], cs);     // accumulate over i across blocks
  }
}

// ---------------------------------------------------------------------------
// Kernel 3: BN2 + posterior gate + padded softmax over L.  One 1024-thread WG.
// ---------------------------------------------------------------------------
__global__ __launch_bounds__(1024) void gate_kernel(
    const float* __restrict__ wsum, const unsigned char* __restrict__ bx,
    const float* __restrict__ g2, const float* __restrict__ b2,
    const float* __restrict__ Wp, const float* __restrict__ bp,
    float* __restrict__ w_out) {
  __shared__ float red[1024];
  __shared__ float gl[B_ * L_];
  __shared__ float mh[H_];
  __shared__ float ivh[H_];
  const int tid = threadIdx.x;

  // BN2 stats per head over (B, L)
  for (int h = 0; h < H_; ++h) {
    float s1 = 0.f, s2 = 0.f;
    for (int idx = tid; idx < B_ * L_; idx += 1024) {
      int bb = idx >> 10, l = idx & (L_ - 1);
      float v = wsum[(size_t)(bb * H_ + h) * L_ + l];
      s1 += v;
      s2 = fmaf(v, v, s2);
    }
    red[tid] = s1; __syncthreads();
    for (int s = 512; s > 0; s >>= 1) { if (tid < s) red[tid] += red[tid + s]; __syncthreads(); }
    float S1 = red[0]; __syncthreads();
    red[tid] = s2; __syncthreads();
    for (int s = 512; s > 0; s >>= 1) { if (tid < s) red[tid] += red[tid + s]; __syncthreads(); }
    float S2 = red[0]; __syncthreads();
    if (tid == 0) {
      const float cnt = (float)(B_ * L_);
      float mean = S1 / cnt;
      float var  = S2 / cnt - mean * mean;
      mh[h]  = mean;
      ivh[h] = rsqrtf(fmaxf(var, 0.f) + EPS_);
    }
    __syncthreads();
  }

  // gate: Linear(H,2) + 2-way softmax, class-0 prob; -inf at padding
  for (int idx = tid; idx < B_ * L_; idx += 1024) {
    int bb = idx >> 10, l = idx & (L_ - 1);
    float z0 = bp[0], z1 = bp[1];
#pragma unroll
    for (int h = 0; h < H_; ++h) {
      float wn = fmaf((wsum[(size_t)(bb * H_ + h) * L_ + l] - mh[h]) * ivh[h], g2[h], b2[h]);
      z0 = fmaf(Wp[h], wn, z0);
      z1 = fmaf(Wp[H_ + h], wn, z1);
    }
    float m = fmaxf(z0, z1);
    float e0 = __expf(z0 - m), e1 = __expf(z1 - m);
    float p0 = e0 / (e0 + e1);
    if (isnan(p0)) p0 = 0.f;
    gl[idx] = bx[idx] ? -INFINITY : p0;
  }
  __syncthreads();

  // final softmax over L per batch
  for (int bb = 0; bb < B_; ++bb) {
    float xv = gl[bb * L_ + tid];
    red[tid] = xv; __syncthreads();
    for (int s = 512; s > 0; s >>= 1) { if (tid < s) red[tid] = fmaxf(red[tid], red[tid + s]); __syncthreads(); }
    float bm = red[0]; __syncthreads();
    float e = __expf(xv - bm);
    red[tid] = e; __syncthreads();
    for (int s = 512; s > 0; s >>= 1) { if (tid < s) red[tid] += red[tid + s]; __syncthreads(); }
    float bs = red[0]; __syncthreads();
    float p = e / bs;
    if (isnan(p)) p = 0.f;
    w_out[bb * L_ + tid] = p;
  }
}

// ---------------------------------------------------------------------------
// Kernel 4: Vh = split_heads(V) permute copy (coalesced, streaming NT both ways)
// ---------------------------------------------------------------------------
__global__ __launch_bounds__(256) void vperm_kernel(const float* __restrict__ V,
                                                    float* __restrict__ vh_out) {
  size_t n = (size_t)blockIdx.x * blockDim.x + threadIdx.x;
  if (n >= (size_t)B_ * H_ * L_ * DH_) return;
  int d = (int)(n & (DH_ - 1));
  size_t r = n >> 7;
  int l = (int)(r & (L_ - 1));
  size_t r2 = r >> 10;
  int h = (int)(r2 % H_);
  int b = (int)(r2 / H_);
  float v = __builtin_nontemporal_load(&V[(size_t)(b * L_ + l) * D_ + h * DH_ + d]);
  __builtin_nontemporal_store(v, &vh_out[n]);
}

// ---------------------------------------------------------------------------
extern "C" void kernel_launch(void* const* d_in, const int* in_sizes, int n_in,
                              void* d_out, int out_size, void* d_ws, size_t ws_size,
                              hipStream_t stream) {
  const float* Q = (const float*)d_in[0];
  const float* K = (const float*)d_in[1];
  const float* V = (const float*)d_in[2];
  // d_in[3] = pad_mask (unused: derivable from bx_packed)
  const unsigned char* bx = (const unsigned char*)d_in[4];
  const float* g1 = (const float*)d_in[5];
  const float* b1 = (const float*)d_in[6];
  const float* g2 = (const float*)d_in[7];
  const float* b2 = (const float*)d_in[8];
  const float* Wp = (const float*)d_in[9];
  const float* bp = (const float*)d_in[10];

  float* ws    = (float*)d_ws;
  float* q2    = ws + WS_Q2;
  float* k2    = ws + WS_K2;
  float* stats = ws + WS_STATS;
  float* meanv = ws + WS_MEAN;
  float* invs  = ws + WS_INVSTD;
  float* wsum  = ws + WS_WSUM;

  float* co_out = (float*)d_out;                              // (B,H,L,L)
  float* w_out  = co_out + (size_t)B_ * H_ * L_ * L_;         // (B,L,1)
  float* vh_out = w_out + (size_t)B_ * L_;                    // (B,H,L,DH)

  // zero the accumulators each launch (graph-capture safe)
  hipMemsetAsync(stats, 0, (size_t)(4 * H_ + B_ * H_ * L_) * sizeof(float), stream);

  sq_kernel<<<(B_ * H_ * L_ + 255) / 256, 256, 0, stream>>>(Q, K, q2, k2);
  stats_kernel<<<B_ * H_ * (L_ / 16), 256, 0, stream>>>(Q, K, q2, k2, stats);
  stats_final<<<1, 32, 0, stream>>>(stats, meanv, invs);
  co_softmax_kernel<<<B_ * H_ * (L_ / 16), 256, 0, stream>>>(
      Q, K, q2, k2, meanv, invs, g1, b1, bx, co_out, wsum);
  gate_kernel<<<1, 1024, 0, stream>>>(wsum, bx, g2, b2, Wp, bp, w_out);
  vperm_kernel<<<(B_ * H_ * L_ * DH_ + 255) / 256, 256, 0, stream>>>(V, vh_out);
}